// Net_14147622273472
// MI455X (gfx1250) — compile-verified
//
#include <hip/hip_runtime.h>
#include <hip/hip_bf16.h>
#include <math.h>

#define N_NODES   100000
#define N_EDGES   3200000
#define N_FEAT    512
#define HIDDEN    16
#define N_CLASSES 8

typedef __attribute__((ext_vector_type(2))) float v2f;
typedef __attribute__((ext_vector_type(8))) float v8f;

// Native L2 fp32 atomic add, no return (STOREcnt-tracked), device scope so the
// RMW resolves at the device-common L2 atomic units.
__device__ __forceinline__ void atomAddF32(float* p, float v) {
    asm volatile("global_atomic_add_f32 %0, %1, off scope:SCOPE_DEV"
                 :: "v"(p), "v"(v) : "memory");
}

// ---------------- normalization ----------------

__global__ void k_deg_init(float* __restrict__ deg) {
    int i = blockIdx.x * blockDim.x + threadIdx.x;
    if (i < N_NODES) deg[i] = 1.0f;   // self-loop weight
}

__global__ void k_deg_accum(float* __restrict__ deg, const int* __restrict__ dst,
                            const float* __restrict__ w) {
    int e = blockIdx.x * blockDim.x + threadIdx.x;
    if (e < N_EDGES) atomAddF32(&deg[dst[e]], w[e]);
}

__global__ void k_dis(float* __restrict__ deg) {
    int i = blockIdx.x * blockDim.x + threadIdx.x;
    if (i < N_NODES) {
        float d = deg[i];
        deg[i] = (d > 0.0f) ? rsqrtf(d) : 0.0f;   // in place: deg -> deg^{-1/2}
    }
}

__global__ void k_norm(float* __restrict__ norm, const float* __restrict__ dis,
                       const int* __restrict__ src, const int* __restrict__ dst,
                       const float* __restrict__ w) {
    int e = blockIdx.x * blockDim.x + threadIdx.x;
    if (e < N_EDGES) norm[e] = dis[src[e]] * w[e] * dis[dst[e]];
}

// ---------------- GEMM 1: [100000x512] @ [512x16], fp32 WMMA ----------------
// One wave per 16-row tile; 6250 tiles = 625 blocks x 10 waves, every wave
// full so EXEC is all ones for each WMMA. Epilogue also writes the self-loop
// message agg1 = h1 * dis^2 (fused k_agg_init16).

__global__ __launch_bounds__(320) void k_gemm1(const float* __restrict__ X,
                                               const float* __restrict__ W,
                                               const float* __restrict__ dis,
                                               float* __restrict__ H,
                                               float* __restrict__ A1) {
    __shared__ float sW[N_FEAT * HIDDEN];          // 32 KB
    for (int i = threadIdx.x; i < N_FEAT * HIDDEN; i += 320) sW[i] = W[i];
    __syncthreads();

    int wave = (blockIdx.x * 320 + (int)threadIdx.x) >> 5;
    int lane = threadIdx.x & 31;
    int m    = lane & 15;            // N index (and A-row index)
    int kh   = (lane >> 4) << 1;     // 0 for lanes 0-15, 2 for lanes 16-31
    int row0 = wave << 4;

    const float* xrow = X + (size_t)(row0 + m) * N_FEAT;
    v8f acc = {0.f, 0.f, 0.f, 0.f, 0.f, 0.f, 0.f, 0.f};

#pragma unroll 4
    for (int k = 0; k < N_FEAT; k += 4) {
        // X is streamed exactly once (~205 MB ~ L2 size): non-temporal so it
        // doesn't evict the reused edge-index / norm / h arrays.
        v2f a = __builtin_nontemporal_load((const v2f*)(xrow + k + kh));
        v2f b;
        b.x = sW[(k + kh) * HIDDEN + m];                      // B row K=kh
        b.y = sW[(k + kh + 1) * HIDDEN + m];                  // B row K=kh+1
        acc = __builtin_amdgcn_wmma_f32_16x16x4_f32(
            false, a, false, b, (short)0, acc, false, false);
    }

    int mb = row0 + ((lane >> 4) << 3);   // lanes>=16 hold M = v+8
#pragma unroll
    for (int v = 0; v < 8; ++v) {
        int   row = mb + v;
        float d   = dis[row];
        H [(size_t)row * HIDDEN + m] = acc[v];
        A1[(size_t)row * HIDDEN + m] = acc[v] * d * d;   // self-loop init
    }
}

// ---------------- layer-1 edge scatter ----------------

__global__ void k_agg_edge16(float* __restrict__ out, const float* __restrict__ h,
                             const float* __restrict__ norm,
                             const int* __restrict__ src, const int* __restrict__ dst) {
    int t = blockIdx.x * blockDim.x + threadIdx.x;   // t < N_EDGES*16
    int e = t >> 4;
    int f = t & 15;
    atomAddF32(&out[dst[e] * HIDDEN + f], h[src[e] * HIDDEN + f] * norm[e]);
}

// ---------------- GEMM 2: [100000x16] @ [16x8] (N padded to 16) -------------
// A-load applies bias + ReLU on the fly (fused k_relu_bias16); epilogue
// writes h2 and initializes d_out = h2 * dis^2 (fused k_agg_init8).

__global__ __launch_bounds__(320) void k_gemm2(const float* __restrict__ Ain,
                                               const float* __restrict__ W2,
                                               const float* __restrict__ b1,
                                               const float* __restrict__ dis,
                                               float* __restrict__ H2,
                                               float* __restrict__ Out) {
    __shared__ float sW[HIDDEN * 16];                // 16x16, cols 8..15 zero
    __shared__ float sB[HIDDEN];
    for (int i = threadIdx.x; i < HIDDEN * 16; i += 320) {
        int r = i >> 4, c = i & 15;
        sW[i] = (c < N_CLASSES) ? W2[r * N_CLASSES + c] : 0.0f;
    }
    if (threadIdx.x < HIDDEN) sB[threadIdx.x] = b1[threadIdx.x];
    __syncthreads();

    int wave = (blockIdx.x * 320 + (int)threadIdx.x) >> 5;
    int lane = threadIdx.x & 31;
    int m    = lane & 15;
    int kh   = (lane >> 4) << 1;
    int row0 = wave << 4;

    const float* arow = Ain + (size_t)(row0 + m) * HIDDEN;
    v8f acc = {0.f, 0.f, 0.f, 0.f, 0.f, 0.f, 0.f, 0.f};

#pragma unroll
    for (int k = 0; k < HIDDEN; k += 4) {
        int kk = k + kh;
        v2f a, b;
        a.x = fmaxf(arow[kk]     + sB[kk],     0.0f);   // bias + ReLU fused
        a.y = fmaxf(arow[kk + 1] + sB[kk + 1], 0.0f);
        b.x = sW[kk * 16 + m];
        b.y = sW[(kk + 1) * 16 + m];
        acc = __builtin_amdgcn_wmma_f32_16x16x4_f32(
            false, a, false, b, (short)0, acc, false, false);
    }

    if (m < N_CLASSES) {
        int mb = row0 + ((lane >> 4) << 3);
#pragma unroll
        for (int v = 0; v < 8; ++v) {
            int   row = mb + v;
            float d   = dis[row];
            H2 [(size_t)row * N_CLASSES + m] = acc[v];
            Out[(size_t)row * N_CLASSES + m] = acc[v] * d * d;  // self-loop init
        }
    }
}

// ---------------- layer-2 edge scatter + log_softmax ----------------

__global__ void k_agg_edge8(float* __restrict__ out, const float* __restrict__ h,
                            const float* __restrict__ norm,
                            const int* __restrict__ src, const int* __restrict__ dst) {
    int t = blockIdx.x * blockDim.x + threadIdx.x;   // t < N_EDGES*8
    int e = t >> 3;
    int f = t & 7;
    atomAddF32(&out[dst[e] * N_CLASSES + f], h[src[e] * N_CLASSES + f] * norm[e]);
}

__global__ void k_logsoftmax(float* __restrict__ out, const float* __restrict__ b2) {
    int i = blockIdx.x * blockDim.x + threadIdx.x;
    if (i >= N_NODES) return;
    float z[N_CLASSES];
    float mx = -INFINITY;
#pragma unroll
    for (int f = 0; f < N_CLASSES; ++f) {
        z[f] = out[i * N_CLASSES + f] + b2[f];
        mx = fmaxf(mx, z[f]);
    }
    float s = 0.0f;
#pragma unroll
    for (int f = 0; f < N_CLASSES; ++f) s += expf(z[f] - mx);
    float lse = mx + logf(s);
#pragma unroll
    for (int f = 0; f < N_CLASSES; ++f) out[i * N_CLASSES + f] = z[f] - lse;
}

// ---------------- launch ----------------

extern "C" void kernel_launch(void* const* d_in, const int* in_sizes, int n_in,
                              void* d_out, int out_size, void* d_ws, size_t ws_size,
                              hipStream_t stream) {
    const float* x  = (const float*)d_in[0];
    const int*   ei = (const int*)d_in[1];        // [2, N_EDGES]
    const float* ew = (const float*)d_in[2];
    const float* W1 = (const float*)d_in[3];
    const float* b1 = (const float*)d_in[4];
    const float* W2 = (const float*)d_in[5];
    const float* b2 = (const float*)d_in[6];
    const int* src = ei;
    const int* dst = ei + N_EDGES;

    float* ws   = (float*)d_ws;
    float* dis  = ws;                                   // N_NODES (deg -> deg^-1/2)
    float* norm = dis + N_NODES;                        // N_EDGES
    float* h1   = norm + N_EDGES;                       // N_NODES*16
    float* agg1 = h1 + (size_t)N_NODES * HIDDEN;        // N_NODES*16
    float* h2   = agg1 + (size_t)N_NODES * HIDDEN;      // N_NODES*8
    float* out  = (float*)d_out;                        // N_NODES*8

    k_deg_init   <<<(N_NODES + 255) / 256, 256, 0, stream>>>(dis);
    k_deg_accum  <<<N_EDGES / 256, 256, 0, stream>>>(dis, dst, ew);
    k_dis        <<<(N_NODES + 255) / 256, 256, 0, stream>>>(dis);
    k_norm       <<<N_EDGES / 256, 256, 0, stream>>>(norm, dis, src, dst, ew);

    k_gemm1      <<<625, 320, 0, stream>>>(x, W1, dis, h1, agg1);
    k_agg_edge16 <<<(N_EDGES / 256) * HIDDEN, 256, 0, stream>>>(agg1, h1, norm, src, dst);

    k_gemm2      <<<625, 320, 0, stream>>>(agg1, W2, b1, dis, h2, out);
    k_agg_edge8  <<<(N_EDGES / 256) * N_CLASSES, 256, 0, stream>>>(out, h2, norm, src, dst);
    k_logsoftmax <<<(N_NODES + 255) / 256, 256, 0, stream>>>(out, b2);
}